// GraphConvolution_18339510354492
// MI455X (gfx1250) — compile-verified
//
#include <hip/hip_runtime.h>
#include <hip/hip_bf16.h>

// GCN layer: support = x@W + b  (WMMA f32 16x16x4, LDS-staged A, pre-packed B),
// then out = scatter-add over edges of (vals*dinv[row]*dinv[col]) * support[col].

typedef __attribute__((ext_vector_type(2))) float v2f;
typedef __attribute__((ext_vector_type(8))) float v8f;

#define D_IN   256
#define D_OUT  128
#define LDS_PITCH 260   // 256 + 4: row stride = 4 banks mod 64 -> 16 rows hit 16 banks

// ---------------- zero fill ----------------
__global__ void gcn_zero_f32(float* __restrict__ p, size_t n) {
    size_t i = (size_t)blockIdx.x * blockDim.x + threadIdx.x;
    if (i < n) p[i] = 0.0f;
}

// ---------------- degree accumulation ----------------
__global__ void gcn_degree(const int* __restrict__ row,
                           const float* __restrict__ vals,
                           float* __restrict__ deg, int E) {
    int e = blockIdx.x * blockDim.x + threadIdx.x;
    if (e < E) atomicAdd(&deg[row[e]], vals[e]);
}

// ---------------- deg -> d^-1/2 (in place) ----------------
__global__ void gcn_rsqrt(float* __restrict__ deg, int n) {
    int i = blockIdx.x * blockDim.x + threadIdx.x;
    if (i < n) {
        float d = deg[i];
        deg[i] = (d > 0.0f) ? rsqrtf(fmaxf(d, 1e-12f)) : 0.0f;
    }
}

// ---------------- pack W into per-lane WMMA B-fragment order ----------------
// Wp[((tile*64 + k0/4)*32 + lane)*2 + j] = W[(k0 + (lane>>4)*2 + j)*128 + tile*16 + (lane&15)]
// so the GEMM reads one contiguous b64 per lane per k-step.
__global__ void gcn_pack_w(const float* __restrict__ W, float* __restrict__ Wp) {
    int idx  = blockIdx.x * blockDim.x + threadIdx.x;   // 0 .. 8*64*32-1
    int lane = idx & 31;
    int k4   = (idx >> 5) & 63;
    int tile = idx >> 11;
    int t    = lane & 15;
    int k    = k4 * 4 + ((lane >> 4) << 1);
    size_t dst = ((size_t)(tile * 64 + k4) * 32 + lane) * 2;
    Wp[dst + 0] = W[(size_t)(k + 0) * D_OUT + tile * 16 + t];
    Wp[dst + 1] = W[(size_t)(k + 1) * D_OUT + tile * 16 + t];
}

// ---------------- support = x @ W + b via V_WMMA_F32_16X16X4_F32 ----------------
// Block = 256 threads = 8 waves; all waves share one 16-row x tile (staged in LDS),
// wave w computes output cols [w*16, +16).
__global__ __launch_bounds__(256) void gcn_gemm_wmma(
    const float* __restrict__ x, const float* __restrict__ Wp,
    const float* __restrict__ b, float* __restrict__ support, int N) {
    __shared__ float lds_x[16 * LDS_PITCH];

    const int tid  = threadIdx.x;
    const int wave = tid >> 5;
    const int lane = tid & 31;
    const int m0   = blockIdx.x * 16;        // tile row base
    const int n0   = wave * 16;              // tile col base (8 waves * 16 = 128)
    const int t    = lane & 15;              // M for A, N for B/C
    const int kh   = (lane >> 4) << 1;       // 0 or 2

    // --- cooperative stage of the 16x256 x tile (4 float4 per thread, coalesced)
    #pragma unroll
    for (int it = 0; it < 4; ++it) {
        int idx = it * 256 + tid;            // 0..1023
        int r   = idx >> 6;                  // 0..15
        int c4  = idx & 63;                  // float4 column
        int mr  = m0 + r;
        if (mr >= N) mr = N - 1;             // clamp: address-safe, EXEC untouched
        float4 v = ((const float4*)(x + (size_t)mr * D_IN))[c4];
        *(float4*)&lds_x[r * LDS_PITCH + c4 * 4] = v;
    }
    __syncthreads();

    const float* __restrict__ arow = &lds_x[t * LDS_PITCH + kh];
    const float* __restrict__ bbase = Wp + ((size_t)wave * 64 * 32 + lane) * 2;

    v8f acc = {};
    #pragma unroll 4
    for (int k0 = 0; k0 < D_IN; k0 += 4) {
        v2f a  = *(const v2f*)(arow + k0);                    // ds_load_b64, bank-clean
        v2f bm = *(const v2f*)(bbase + (size_t)(k0 >> 2) * 64); // global_load_b64, coalesced
        // 8 args: (neg_a, A, neg_b, B, c_mod, C, reuse_a, reuse_b)
        acc = __builtin_amdgcn_wmma_f32_16x16x4_f32(
            false, a, false, bm, (short)0, acc, false, false);
    }

    // write back D + bias.  VGPR v: lanes 0-15 -> M=v; lanes 16-31 -> M=8+v.
    const int n     = lane & 15;
    const int mbase = m0 + ((lane >> 4) << 3);
    const float bias = b[n0 + n];
    float* __restrict__ dst = support + (size_t)mbase * D_OUT + n0 + n;
    if (m0 + 16 <= N) {                      // uniform fast path: straight-line stores
        #pragma unroll
        for (int v = 0; v < 8; ++v) dst[(size_t)v * D_OUT] = acc[v] + bias;
    } else {                                 // cold tail, per-row predication
        #pragma unroll
        for (int v = 0; v < 8; ++v)
            if (mbase + v < N) dst[(size_t)v * D_OUT] = acc[v] + bias;
    }
}

// ---------------- SpMM scatter: one wave per edge ----------------
__global__ __launch_bounds__(256) void gcn_spmm(
    const int* __restrict__ row, const int* __restrict__ col,
    const float* __restrict__ vals, const float* __restrict__ dinv,
    const float* __restrict__ support, float* __restrict__ out, int E) {
    const int e = blockIdx.x * 8 + (threadIdx.x >> 5);
    if (e >= E) return;
    const int lane = threadIdx.x & 31;
    const int r = row[e];
    const int c = col[e];
    const float w = vals[e] * dinv[r] * dinv[c];

    const float4 v = ((const float4*)(support + (size_t)c * D_OUT))[lane];
    float* __restrict__ dst = out + (size_t)r * D_OUT + lane * 4;
    atomicAdd(dst + 0, w * v.x);
    atomicAdd(dst + 1, w * v.y);
    atomicAdd(dst + 2, w * v.z);
    atomicAdd(dst + 3, w * v.w);
}

extern "C" void kernel_launch(void* const* d_in, const int* in_sizes, int n_in,
                              void* d_out, int out_size, void* d_ws, size_t ws_size,
                              hipStream_t stream) {
    const float* x    = (const float*)d_in[0];   // [N, 256]
    const int*   row  = (const int*)  d_in[1];   // [E]
    const int*   col  = (const int*)  d_in[2];   // [E]
    const float* vals = (const float*)d_in[3];   // [E]
    const float* W    = (const float*)d_in[4];   // [256, 128]
    const float* b    = (const float*)d_in[5];   // [128]
    float*       out  = (float*)d_out;           // [N, 128]

    const int N = in_sizes[0] / D_IN;
    const int E = in_sizes[1];

    // workspace: [deg/dinv : N f32][support : N*128 f32][Wp : 32768 f32]
    float* deg = (float*)d_ws;
    size_t off = ((size_t)N * sizeof(float) + 255) & ~(size_t)255;
    float* support = (float*)((char*)d_ws + off);
    off += (((size_t)N * D_OUT * sizeof(float)) + 255) & ~(size_t)255;
    float* Wp = (float*)((char*)d_ws + off);

    // 1) zero deg, 2) zero out
    gcn_zero_f32<<<(N + 255) / 256, 256, 0, stream>>>(deg, (size_t)N);
    size_t out_n = (size_t)N * D_OUT;
    gcn_zero_f32<<<(unsigned)((out_n + 255) / 256), 256, 0, stream>>>(out, out_n);

    // 3) degrees, 4) d^-1/2, 5) pack W into fragment order
    gcn_degree<<<(E + 255) / 256, 256, 0, stream>>>(row, vals, deg, E);
    gcn_rsqrt<<<(N + 255) / 256, 256, 0, stream>>>(deg, N);
    gcn_pack_w<<<(8 * 64 * 32) / 256, 256, 0, stream>>>(W, Wp);

    // 6) support = x@W + b  (one 16-row tile per block, 8 waves cover 128 cols)
    gcn_gemm_wmma<<<(N + 15) / 16, 256, 0, stream>>>(x, Wp, b, support, N);

    // 7) edge scatter (8 edges per 256-thread block, one wave per edge)
    gcn_spmm<<<(E + 7) / 8, 256, 0, stream>>>(row, col, vals, deg, support, out, E);
}